// ROIPooling_7215545057804
// MI455X (gfx1250) — compile-verified
//
#include <hip/hip_runtime.h>
#include <hip/hip_bf16.h>
#include <cstdint>

// Problem geometry (fixed by the reference):
//   x: (2048, 256, 256) f32;  crop 43x43;  window 37x37 stride 1 -> out (2048,7,7)
#define GRID_N   7
#define WIN      37          // round(256/7)
#define CROP     43          // GRID_N + WIN - 1
#define IMG      256
#define TSTRIDE  44          // LDS row stride in floats: 176 B, 16B-aligned for B128
#define SEGS     11          // 16-byte segments per row (44 floats)
#define NTRANS   (CROP * SEGS)   // 473 async B128 transfers per channel
#define BLOCK    512

typedef int v4i __attribute__((ext_vector_type(4)));
typedef __attribute__((address_space(1))) v4i* as1_v4i_p;   // global (prints as __device__)
typedef __attribute__((address_space(3))) v4i* as3_v4i_p;   // LDS

// Async 16B global -> LDS copy (gfx1250 GLOBAL_LOAD_ASYNC_TO_LDS_B128, ASYNCcnt).
__device__ __forceinline__ void async_copy16(const float* g, float* l) {
#if __has_builtin(__builtin_amdgcn_global_load_async_to_lds_b128)
    __builtin_amdgcn_global_load_async_to_lds_b128((as1_v4i_p)g, (as3_v4i_p)l,
                                                   /*offset=*/0, /*cpol=*/0);
#else
    unsigned loff = (unsigned)(unsigned long long)(as3_v4i_p)l;
    asm volatile("global_load_async_to_lds_b128 %0, %1, off"
                 :: "v"(loff), "v"(g) : "memory");
#endif
}

__device__ __forceinline__ void async_wait0() {
#if __has_builtin(__builtin_amdgcn_s_wait_asynccnt)
    __builtin_amdgcn_s_wait_asynccnt(0);
#else
    asm volatile("s_wait_asynccnt 0x0" ::: "memory");
#endif
}

__global__ __launch_bounds__(BLOCK)
void roi_maxpool_kernel(const float* __restrict__ x, float* __restrict__ out) {
    __shared__ float tile[CROP * TSTRIDE];      // 43 x 44 floats (7568 B)
    __shared__ float rowmax[CROP * GRID_N];     // 43 x 7  floats

    const int c   = blockIdx.x;
    const int tid = threadIdx.x;
    const float* src = x + (size_t)c * (IMG * IMG);

    // ---- Stage the 43x43 tile into LDS via async DMA (one B128 per lane) ----
    if (tid < NTRANS) {
        const int row = tid / SEGS;              // 0..42
        const int seg = tid - row * SEGS;        // 0..10  (44 floats >= 43 needed)
        const float* g = src + row * IMG + seg * 4;          // 16B-aligned (row = 1KB)
        float*       l = &tile[row * TSTRIDE + seg * 4];     // 16B-aligned (176B rows)
        async_copy16(g, l);
    }
    async_wait0();        // drain this wave's ASYNCcnt (LDS writes visible)
    __syncthreads();      // make all waves' tiles visible to everyone

    // ---- Phase 1: row-wise sliding max. thread = (r, j), 43*7 = 301 threads ----
    if (tid < CROP * GRID_N) {
        const int r = tid / GRID_N;
        const int j = tid - r * GRID_N;
        const float* rp = &tile[r * TSTRIDE + j];
        float m = rp[0];
        #pragma unroll
        for (int k = 1; k < WIN; ++k) m = fmaxf(m, rp[k]);
        rowmax[tid] = m;                         // layout [r][j]
    }
    __syncthreads();

    // ---- Phase 2: column-wise sliding max over rowmax. 7*7 = 49 threads ----
    if (tid < GRID_N * GRID_N) {
        const int i = tid / GRID_N;
        const int j = tid - i * GRID_N;
        float m = rowmax[i * GRID_N + j];
        #pragma unroll
        for (int r = 1; r < WIN; ++r) m = fmaxf(m, rowmax[(i + r) * GRID_N + j]);
        out[(size_t)c * (GRID_N * GRID_N) + tid] = m;
    }
}

extern "C" void kernel_launch(void* const* d_in, const int* in_sizes, int n_in,
                              void* d_out, int out_size, void* d_ws, size_t ws_size,
                              hipStream_t stream) {
    const float* x   = (const float*)d_in[0];
    float*       out = (float*)d_out;
    const int channels = in_sizes[0] / (IMG * IMG);   // 2048
    roi_maxpool_kernel<<<dim3(channels), dim3(BLOCK), 0, stream>>>(x, out);
}